// MarginLoss_70102456205378
// MI455X (gfx1250) — compile-verified
//
#include <hip/hip_runtime.h>
#include <cstdint>
#include <cstddef>

#define NCOL 64
#define RPB  128            // rows per block == threads per block (4 waves)
#define RSTRIDE 272         // LDS row stride in bytes: 68 dwords, 16B-aligned, spreads banks

typedef float v2f __attribute__((ext_vector_type(2)));
typedef float v8f __attribute__((ext_vector_type(8)));

// Compile-time coefficients:
//   w[i]  = 1/(B*(n-i)) for lag i in 1..62
//   wh[i] = w[i]/2                       (weight of the |d| half)
//   c[m]  = 0.5*(sum_{i<=min(m,62)} w[i] - sum_{i<=min(63-m,62)} w[i])   (linear half)
struct Coefs { float wh[NCOL]; float c[NCOL]; };

constexpr Coefs make_coefs() {
  Coefs co{};
  float w[NCOL] = {};
  const float Bf = 524288.0f;
  for (int i = 1; i <= NCOL - 2; ++i) w[i] = 1.0f / (Bf * (float)(NCOL - i));
  for (int i = 0; i < NCOL; ++i) co.wh[i] = 0.5f * w[i];
  for (int m = 0; m < NCOL; ++m) {
    int ca = (m < NCOL - 2) ? m : (NCOL - 2);
    int cb = ((NCOL - 1 - m) < (NCOL - 2)) ? (NCOL - 1 - m) : (NCOL - 2);
    float a = 0.0f, b = 0.0f;
    for (int i = 1; i <= ca; ++i) a += w[i];
    for (int i = 1; i <= cb; ++i) b += w[i];
    co.c[m] = 0.5f * (a - b);
  }
  return co;
}
__device__ constexpr Coefs COEF = make_coefs();

// Kernel 1: per-row weighted hinge sums -> one partial per block.
__global__ __launch_bounds__(RPB) void hinge_rows_kernel(const float* __restrict__ score,
                                                         float* __restrict__ partials) {
  __shared__ __align__(16) char lbuf[RPB * RSTRIDE];
  __shared__ float red[RPB];

  const int tid = threadIdx.x;
  const float* tile = score + (size_t)blockIdx.x * (size_t)(RPB * NCOL);
  unsigned ldsBase = (unsigned)(uintptr_t)(void*)&lbuf[0];

  // ---- CDNA5 async DMA: stage 128 rows (32 KB) into LDS, fully coalesced ----
  // 2048 x 16B chunks; thread t moves chunks t, t+128, ... (16 per thread).
#pragma unroll
  for (int it = 0; it < (RPB * NCOL * 4) / 16 / RPB; ++it) {
    int q = tid + it * RPB;
    unsigned loff = ldsBase + (unsigned)((q >> 4) * RSTRIDE + (q & 15) * 16);
    const char* g = (const char*)tile + (size_t)q * 16;
    asm volatile("global_load_async_to_lds_b128 %0, %1, off"
                 :: "v"(loff), "v"(g) : "memory");
  }
  asm volatile("s_wait_asynccnt 0" ::: "memory");
  __syncthreads();

  // ---- load own row from LDS, fold margin: u[j] = s[j] + 0.01*j ----
  float u[NCOL];
  const float4* rp = (const float4*)(lbuf + tid * RSTRIDE);
#pragma unroll
  for (int j = 0; j < NCOL / 4; ++j) {
    float4 v = rp[j];
    u[4 * j + 0] = v.x + 0.01f * (float)(4 * j + 0);
    u[4 * j + 1] = v.y + 0.01f * (float)(4 * j + 1);
    u[4 * j + 2] = v.z + 0.01f * (float)(4 * j + 2);
    u[4 * j + 3] = v.w + 0.01f * (float)(4 * j + 3);
  }

  // relu(x) = (x + |x|)/2:  linear half is 64 FMAs with constexpr coefs,
  // abs half is 2 VALU ops per pair (v_sub + v_add with |src| modifier).
  float acc = 0.0f;
#pragma unroll
  for (int m = 0; m < NCOL; ++m) acc = fmaf(u[m], COEF.c[m], acc);

#pragma unroll
  for (int i = 1; i <= NCOL - 2; ++i) {
    float t = 0.0f;
#pragma unroll
    for (int j = 0; j < NCOL - i; ++j) t += fabsf(u[j + i] - u[j]);
    acc = fmaf(t, COEF.wh[i], acc);
  }

  // ---- deterministic block tree reduction ----
  red[tid] = acc;
  __syncthreads();
#pragma unroll
  for (int s = RPB / 2; s > 0; s >>= 1) {
    if (tid < s) red[tid] += red[tid + s];
    __syncthreads();
  }
  if (tid == 0) partials[blockIdx.x] = red[0];
}

// Kernel 2: single wave32; reduce partials with V_WMMA_F32_16X16X4_F32.
// A[m][k] = p[it*64 + m*4 + k], B = ones(4x16) => D[m][n] += rowsum(m).
// REQUIRES: n a multiple of 64 (launcher guarantees 4096).
__global__ __launch_bounds__(32) void reduce_wmma_kernel(const float* __restrict__ p,
                                                         float* __restrict__ out, int n) {
  int lane = threadIdx.x;
  int m = lane & 15;
  int koff = (lane >> 4) << 1;   // lanes 0-15 hold K=0,1; lanes 16-31 hold K=2,3
  v8f acc = {};
  v2f ones; ones.x = 1.0f; ones.y = 1.0f;
  int iters = n >> 6;
  for (int it = 0; it < iters; ++it) {
    int base = it * 64 + m * 4 + koff;
    v2f a = *(const v2f*)(p + base);           // unguarded: merges to global_load_b64
    acc = __builtin_amdgcn_wmma_f32_16x16x4_f32(false, a, false, ones,
                                                (short)0, acc, false, false);
  }
  // lane l (<16) holds D[0..7][l]; lane l+16 holds D[8..15][l]. Sum M within
  // lane, then combine lanes 0 and 16 (column N=0) for the grand total.
  float s = acc[0] + acc[1] + acc[2] + acc[3] + acc[4] + acc[5] + acc[6] + acc[7];
  float tot = __shfl(s, 0, 32) + __shfl(s, 16, 32);
  if (lane == 0) out[0] = tot;
}

extern "C" void kernel_launch(void* const* d_in, const int* in_sizes, int n_in,
                              void* d_out, int out_size, void* d_ws, size_t ws_size,
                              hipStream_t stream) {
  (void)n_in; (void)out_size; (void)ws_size;
  const float* score = (const float*)d_in[0];
  float* out = (float*)d_out;
  float* partials = (float*)d_ws;

  int rows = in_sizes[0] / NCOL;        // 524288
  int nblocks = rows / RPB;             // 4096 partials (16 KB of workspace), 4096 % 64 == 0

  hipLaunchKernelGGL(hinge_rows_kernel, dim3(nblocks), dim3(RPB), 0, stream,
                     score, partials);
  hipLaunchKernelGGL(reduce_wmma_kernel, dim3(1), dim3(32), 0, stream,
                     partials, out, nblocks);
}